// Net_86234353369143
// MI455X (gfx1250) — compile-verified
//
#include <hip/hip_runtime.h>
#include <stdint.h>
#include <math.h>

#define N_NODES   100000
#define N_EDGES   1600000
#define DIM       64
#define NUM_LAYERS 8
#define NH        (N_NODES * DIM)

typedef __attribute__((ext_vector_type(2))) float v2f;
typedef __attribute__((ext_vector_type(8))) float v8f;

// ---------------------------------------------------------------- utilities
__global__ void zero_kernel4(float4* __restrict__ p, int n4) {
    int i = blockIdx.x * blockDim.x + threadIdx.x;
    if (i < n4) p[i] = make_float4(0.f, 0.f, 0.f, 0.f);
}

__global__ void zero_kernel(float* __restrict__ p, int n) {
    int i = blockIdx.x * blockDim.x + threadIdx.x;
    if (i < n) p[i] = 0.0f;
}

// ------------------------------------------------ input GEMM: x = relu(X@W0^T + b0)
// One wave per 16-row block. A = X[16x64] (fp32 WMMA 16x16x4), B[k][n] = W0[n][k].
__global__ __launch_bounds__(32)
void input_gemm(const float* __restrict__ X, const float* __restrict__ W0,
                const float* __restrict__ b0, float* __restrict__ x,
                float* __restrict__ x0) {
    const int lane = threadIdx.x;
    const int m    = lane & 15;
    const int half = lane >> 4;
    const int row0 = blockIdx.x * 16;

    // A fragments: 16 K-chunks of K=4. Lane layout per ISA:
    // lanes 0-15: v0=K0,v1=K1 ; lanes 16-31: v0=K2,v1=K3
    v2f a[16];
    const float* arow = X + (size_t)(row0 + m) * DIM;
#pragma unroll
    for (int kc = 0; kc < 16; ++kc) {
        a[kc] = *(const v2f*)(arow + kc * 4 + half * 2);
    }

#pragma unroll
    for (int n0 = 0; n0 < 4; ++n0) {
        const int ncol = n0 * 16 + m;     // this lane's N index for B fragments
        // Batch-load all B fragments for this column tile first (one wait, no
        // per-WMMA load stalls). B[k][n] = W0[n][k] -> contiguous pairs of row n.
        v2f b[16];
#pragma unroll
        for (int kc = 0; kc < 16; ++kc) {
            b[kc] = *(const v2f*)(W0 + (size_t)ncol * DIM + kc * 4 + half * 2);
        }
        v8f acc0 = {}, acc1 = {};
#pragma unroll
        for (int kc = 0; kc < 16; kc += 2) {
            acc0 = __builtin_amdgcn_wmma_f32_16x16x4_f32(
                false, a[kc],     false, b[kc],     (short)0, acc0, false, false);
            acc1 = __builtin_amdgcn_wmma_f32_16x16x4_f32(
                false, a[kc + 1], false, b[kc + 1], (short)0, acc1, false, false);
        }
        // C/D layout: VGPR v -> M = v + 8*half ; N = lane&15 (+16*n0)
        const float bias = b0[n0 * 16 + m];
#pragma unroll
        for (int v = 0; v < 8; ++v) {
            int row = row0 + v + 8 * half;
            int col = n0 * 16 + m;
            float r = acc0[v] + acc1[v] + bias;
            r = r > 0.0f ? r : 0.0f;
            x [(size_t)row * DIM + col] = r;
            x0[(size_t)row * DIM + col] = r;
        }
    }
}

// ------------------------------------------------ edge scatter: agg[dst] += x[src]
// 16 lanes per edge, float4 per lane => one fully-coalesced 256B gather per edge.
__global__ __launch_bounds__(256)
void scatter_kernel(const long long* __restrict__ ei,
                    const float* __restrict__ x, float* __restrict__ agg) {
    long long gid = (long long)blockIdx.x * blockDim.x + threadIdx.x;
    int e = (int)(gid >> 4);
    int c = ((int)gid & 15) * 4;
    if (e >= N_EDGES) return;
    int s = (int)ei[e];
    int d = (int)ei[N_EDGES + e];
    const float4 v = *(const float4*)(x + (size_t)s * DIM + c);
    float* dst = agg + (size_t)d * DIM + c;
    atomicAdd(dst + 0, v.x);
    atomicAdd(dst + 1, v.y);
    atomicAdd(dst + 2, v.z);
    atomicAdd(dst + 3, v.w);
}

// -------------------- per-layer: h = 0.9*agg + 0.1*x0 ; x = relu((1-b)h + b*(h@Wl))
__global__ __launch_bounds__(32)
void combine_gemm(const float* __restrict__ agg, const float* __restrict__ x0,
                  const float* __restrict__ Wl, float beta,
                  float* __restrict__ xout) {
    const int lane = threadIdx.x;
    const int m    = lane & 15;
    const int half = lane >> 4;
    const int row0 = blockIdx.x * 16;
    const float omb = 1.0f - beta;

    // A fragments built on the fly: h = 0.9*agg + 0.1*x0
    v2f a[16];
    const float* ga = agg + (size_t)(row0 + m) * DIM;
    const float* g0 = x0  + (size_t)(row0 + m) * DIM;
#pragma unroll
    for (int kc = 0; kc < 16; ++kc) {
        v2f va = *(const v2f*)(ga + kc * 4 + half * 2);
        v2f v0 = *(const v2f*)(g0 + kc * 4 + half * 2);
        a[kc].x = 0.9f * va.x + 0.1f * v0.x;
        a[kc].y = 0.9f * va.y + 0.1f * v0.y;
    }

#pragma unroll
    for (int n0 = 0; n0 < 4; ++n0) {
        const int ncol = n0 * 16 + m;
        // Batch-load B fragments: B[k][n] = Wl[k][n] (row-major 64x64)
        v2f b[16];
#pragma unroll
        for (int kc = 0; kc < 16; ++kc) {
            int k = kc * 4 + half * 2;
            b[kc].x = Wl[(size_t)(k    ) * DIM + ncol];
            b[kc].y = Wl[(size_t)(k + 1) * DIM + ncol];
        }
        v8f acc0 = {}, acc1 = {};
#pragma unroll
        for (int kc = 0; kc < 16; kc += 2) {
            acc0 = __builtin_amdgcn_wmma_f32_16x16x4_f32(
                false, a[kc],     false, b[kc],     (short)0, acc0, false, false);
            acc1 = __builtin_amdgcn_wmma_f32_16x16x4_f32(
                false, a[kc + 1], false, b[kc + 1], (short)0, acc1, false, false);
        }
#pragma unroll
        for (int v = 0; v < 8; ++v) {
            int row = row0 + v + 8 * half;
            int col = n0 * 16 + m;
            size_t idx = (size_t)row * DIM + col;
            float h = 0.9f * agg[idx] + 0.1f * x0[idx];
            float r = omb * h + beta * (acc0[v] + acc1[v]);
            xout[idx] = r > 0.0f ? r : 0.0f;
        }
    }
}

// ---------------------------------------------------------------- column sum of x
__global__ __launch_bounds__(256)
void colsum_kernel(const float* __restrict__ x, float* __restrict__ cs) {
    int g = blockIdx.x * blockDim.x + threadIdx.x;
    int col = g & 63;
    int row = g >> 6;
    int rstride = (gridDim.x * blockDim.x) >> 6;
    float s = 0.0f;
    for (int r = row; r < N_NODES; r += rstride)
        s += x[(size_t)r * DIM + col];
    atomicAdd(&cs[col], s);
}

// -------------------------------- out[j] = b1[j] + sum_k mean_x[k] * W1[j][k]
__global__ __launch_bounds__(64)
void final_kernel(const float* __restrict__ cs, const float* __restrict__ W1,
                  const float* __restrict__ b1, float* __restrict__ out) {
    int j = threadIdx.x;
    if (j < DIM) {
        const float invn = 1.0f / (float)N_NODES;
        float s = 0.0f;
#pragma unroll
        for (int k = 0; k < DIM; ++k)
            s += (cs[k] * invn) * W1[(size_t)j * DIM + k];
        out[j] = s + b1[j];
    }
}

// ---------------------------------------------------------------------------
extern "C" void kernel_launch(void* const* d_in, const int* in_sizes, int n_in,
                              void* d_out, int out_size, void* d_ws, size_t ws_size,
                              hipStream_t stream) {
    (void)in_sizes; (void)n_in; (void)out_size; (void)ws_size;

    const float*     X     = (const float*)d_in[0];
    const long long* ei    = (const long long*)d_in[1];   // int64 edge_index [2, E]
    const float*     W0    = (const float*)d_in[2];
    const float*     b0    = (const float*)d_in[3];
    const float*     W1    = (const float*)d_in[4];
    const float*     b1    = (const float*)d_in[5];
    const float*     convw = (const float*)d_in[6];
    float*           out   = (float*)d_out;

    float* x   = (float*)d_ws;       // [N, 64]
    float* x0  = x  + NH;            // [N, 64]
    float* agg = x0 + NH;            // [N, 64]
    float* cs  = agg + NH;           // [64]

    const int ROWBLOCKS = N_NODES / 16;            // 6250, exact
    input_gemm<<<ROWBLOCKS, 32, 0, stream>>>(X, W0, b0, x, x0);

    for (int l = 0; l < NUM_LAYERS; ++l) {
        zero_kernel4<<<(NH / 4 + 255) / 256, 256, 0, stream>>>((float4*)agg, NH / 4);
        scatter_kernel<<<(int)(((long long)N_EDGES * 16) / 256), 256, 0, stream>>>(ei, x, agg);
        float beta = logf(0.5f / (float)(l + 1) + 1.0f);
        combine_gemm<<<ROWBLOCKS, 32, 0, stream>>>(agg, x0,
                                                   convw + (size_t)l * DIM * DIM,
                                                   beta, x);
    }

    zero_kernel<<<1, 64, 0, stream>>>(cs, 64);
    colsum_kernel<<<256, 256, 0, stream>>>(x, cs);
    final_kernel<<<1, 64, 0, stream>>>(cs, W1, b1, out);
}